// Net_46858093199675
// MI455X (gfx1250) — compile-verified
//
#include <hip/hip_runtime.h>
#include <hip/hip_bf16.h>

typedef __attribute__((ext_vector_type(2))) float v2f;
typedef __attribute__((ext_vector_type(8))) float v8f;

// ---------------------------------------------------------------------------
// Utility: fill a float buffer with a constant (re-init accumulators/degrees
// every call -> deterministic under graph replay).
// ---------------------------------------------------------------------------
__global__ void fill_f32_kernel(float* __restrict__ p, float v, int n) {
    int i = blockIdx.x * blockDim.x + threadIdx.x;
    if (i < n) p[i] = v;
}

// ---------------------------------------------------------------------------
// In-degree accumulation: deg[dst[e]] += 1 (deg pre-filled with 1.0 = self loop).
// ---------------------------------------------------------------------------
__global__ void deg_kernel(const int* __restrict__ dst, int E,
                           float* __restrict__ deg) {
    int e = blockIdx.x * blockDim.x + threadIdx.x;
    if (e < E) atomicAdd(&deg[dst[e]], 1.0f);
}

__global__ void rsqrt_kernel(float* __restrict__ deg, int n) {
    int i = blockIdx.x * blockDim.x + threadIdx.x;
    if (i < n) {
        float d = deg[i];
        deg[i] = (d > 0.0f) ? rsqrtf(d) : 0.0f;   // dinv in place
    }
}

// ---------------------------------------------------------------------------
// Dense transform C[M,N] = A[M,K] @ B[K,N] with V_WMMA_F32_16X16X4_F32.
//
//   block = 128 threads = 4 wave32.  Wave w owns M-tile rows
//   [ (4*blockIdx.x + w)*16 , +16 ) and iterates ALL NT N-tiles itself, so
//   each 8-byte A load feeds NT WMMAs and A is read from HBM exactly once.
//
//   B is staged zero-padded into LDS ([Kpad][80] f32, stride 80 so the four
//   K-rows read per step map to disjoint 16-bank groups -> conflict-free).
//   Staging uses GLOBAL_LOAD_ASYNC_TO_LDS_B32 (ASYNCcnt) for valid cells and
//   plain DS stores of 0 for pad cells.
//
//   Branch-free inner loop: A loads are unconditional v2f (b64) loads with a
//   clamped address; garbage beyond K multiplies zero-padded B rows, garbage
//   rows beyond M feed D rows that the masked store never writes.
//   Requires K even (true here: 128 and 50).  EXEC is all-1s at every WMMA
//   (only wave-uniform early exits).
// ---------------------------------------------------------------------------
template <int NT>
__global__ void wmma_gemm_kernel(const float* __restrict__ A,
                                 const float* __restrict__ B,
                                 float* __restrict__ C,
                                 int M, int K, int N) {
    constexpr int BSTR = 80;                       // LDS row stride (floats)
    __shared__ float Bs[128 * BSTR];               // Kpad x 80, 40 KB
    const int Kpad = (K + 3) & ~3;                 // multiple of 4, <= 128
    const int NCOL = 16 * NT;

    // ---- Stage B -> LDS (async copy for valid cells, zeros for padding) ----
    for (int idx = threadIdx.x; idx < Kpad * NCOL; idx += blockDim.x) {
        int k = idx / NCOL;
        int n = idx - k * NCOL;
        float* lp = &Bs[k * BSTR + n];
        if (k < K && n < N) {
            unsigned lds_off = (unsigned)(unsigned long long)lp;     // as3 offset
            unsigned long long gaddr =
                (unsigned long long)(B + (size_t)k * N + n);
            asm volatile("global_load_async_to_lds_b32 %0, %1, off"
                         :: "v"(lds_off), "v"(gaddr) : "memory");
        } else {
            *lp = 0.0f;
        }
    }
    asm volatile("s_wait_asynccnt 0x0" ::: "memory");  // my async writes done
    __syncthreads();                                   // everyone's done

    const int wave   = threadIdx.x >> 5;
    const int lane   = threadIdx.x & 31;
    const int m_base = (blockIdx.x * 4 + wave) * 16;
    if (m_base >= M) return;                       // wave-uniform exit

    const int m   = lane & 15;                     // row within tile / B col
    const int kh  = lane >> 4;                     // K-half select (ISA 7.12.2)
    const int row = m_base + m;
    const int rowc = row < M ? row : M - 1;        // clamp (values ignored)
    const float* __restrict__ Arow = A + (size_t)rowc * K;

    v8f acc[NT];
    v8f zero = {};
#pragma unroll
    for (int nt = 0; nt < NT; ++nt) acc[nt] = zero;

    for (int k0 = 0; k0 < Kpad; k0 += 4) {
        const int ka  = k0 + 2 * kh;
        const int kac = (ka + 1 < K) ? ka : (K - 2);   // clamp addr; K even
        v2f a = *(const v2f*)(Arow + kac);             // unconditional b64
#pragma unroll
        for (int nt = 0; nt < NT; ++nt) {
            v2f b;
            b.x = Bs[(ka)     * BSTR + nt * 16 + m];
            b.y = Bs[(ka + 1) * BSTR + nt * 16 + m];
            acc[nt] = __builtin_amdgcn_wmma_f32_16x16x4_f32(
                          /*neg_a=*/false, a, /*neg_b=*/false, b,
                          /*c_mod=*/(short)0, acc[nt],
                          /*reuse_a=*/false, /*reuse_b=*/false);
        }
    }

    // C/D layout: VGPR r, half kh -> row = r + 8*kh, col = lane[3:0]
#pragma unroll
    for (int nt = 0; nt < NT; ++nt) {
#pragma unroll
        for (int r = 0; r < 8; ++r) {
            int orow = m_base + r + 8 * kh;
            int ocol = nt * 16 + m;
            if (orow < M && ocol < N)
                C[(size_t)orow * N + ocol] = acc[nt][r];
        }
    }
}

// ---------------------------------------------------------------------------
// Edge scatter: acc[dst] += h[src] * dinv[src]*dinv[dst]   (F = 50 or 10)
// One thread per edge; gathers are L2-resident (h fits in 192 MB L2).
// ---------------------------------------------------------------------------
__global__ void scatter_kernel(const float* __restrict__ h,
                               const int* __restrict__ src,
                               const int* __restrict__ dst,
                               const float* __restrict__ dinv,
                               float* __restrict__ acc,
                               int E, int F) {
    int e = blockIdx.x * blockDim.x + threadIdx.x;
    if (e >= E) return;
    const int s = src[e];
    const int d = dst[e];
    const float nrm = dinv[s] * dinv[d];
    const float* __restrict__ hs = h   + (size_t)s * F;
    float* __restrict__       ad = acc + (size_t)d * F;
    __builtin_prefetch(hs, 0, 0);                  // global_prefetch_b8
    for (int f = 0; f < F; ++f) atomicAdd(&ad[f], hs[f] * nrm);
}

// ---------------------------------------------------------------------------
// Epilogue: add self-loop message h[i]*dinv[i]^2, bias, optional ReLU.
// In place into acc (acc becomes the layer output).
// ---------------------------------------------------------------------------
__global__ void finish_kernel(const float* __restrict__ h,
                              const float* __restrict__ dinv,
                              const float* __restrict__ bias,
                              float* __restrict__ acc,
                              int Nn, int F, int do_relu) {
    int i = blockIdx.x * blockDim.x + threadIdx.x;
    int node = i / F;
    if (node >= Nn) return;
    int f = i - node * F;
    float di = dinv[node];
    float v = acc[i] + h[i] * di * di + bias[f];
    if (do_relu) v = fmaxf(v, 0.0f);
    acc[i] = v;
}

// ---------------------------------------------------------------------------
// Decoder: logits[e] = dot(z[src_e], z[dst_e]) over concat(pos, neg) edges.
// F is even (10); rows are 8-byte aligned (stride 40 B) -> float2 loads.
// ---------------------------------------------------------------------------
__global__ void decode_kernel(const float* __restrict__ z,
                              const int* __restrict__ ps, const int* __restrict__ pd,
                              const int* __restrict__ ns, const int* __restrict__ nd,
                              float* __restrict__ out, int EP, int F) {
    int e = blockIdx.x * blockDim.x + threadIdx.x;
    if (e >= 2 * EP) return;
    int s, d;
    if (e < EP) { s = ps[e];      d = pd[e]; }
    else        { s = ns[e - EP]; d = nd[e - EP]; }
    const float2* __restrict__ zs = (const float2*)(z + (size_t)s * F);
    const float2* __restrict__ zd = (const float2*)(z + (size_t)d * F);
    float acc = 0.0f;
    const int F2 = F >> 1;
    for (int f = 0; f < F2; ++f) {
        float2 a = zs[f], b = zd[f];
        acc += a.x * b.x + a.y * b.y;
    }
    out[e] = acc;
}

// ---------------------------------------------------------------------------
extern "C" void kernel_launch(void* const* d_in, const int* in_sizes, int n_in,
                              void* d_out, int out_size, void* d_ws, size_t ws_size,
                              hipStream_t stream) {
    const float* x   = (const float*)d_in[0];
    const float* W1  = (const float*)d_in[1];
    const float* b1  = (const float*)d_in[2];
    const float* W2  = (const float*)d_in[3];
    const float* b2  = (const float*)d_in[4];
    const int*   tei = (const int*)d_in[5];    // [2, E]
    const int*   pei = (const int*)d_in[6];    // [2, EP]
    const int*   nei = (const int*)d_in[7];    // [2, EP]

    const int H1   = in_sizes[2];              // 50
    const int H2   = in_sizes[4];              // 10
    const int F_IN = in_sizes[1] / H1;         // 128
    const int Nn   = in_sizes[0] / F_IN;       // 100000
    const int E    = in_sizes[5] / 2;          // 3.2M
    const int EP   = in_sizes[6] / 2;          // 1.6M

    const int* src = tei;
    const int* dst = tei + E;

    // Workspace carve-up (256B aligned): ~48.4 MB total.
    char* ws = (char*)d_ws;
    size_t off = 0;
    auto carve = [&](size_t bytes) -> float* {
        float* p = (float*)(ws + off);
        off = (off + bytes + 255) & ~(size_t)255;
        return p;
    };
    float* h_lin1 = carve((size_t)Nn * H1 * 4);  // x @ W1
    float* acc1   = carve((size_t)Nn * H1 * 4);  // layer-1 aggregate -> h1
    float* dinv   = carve((size_t)Nn * 4);       // deg -> rsqrt(deg)
    float* h_lin2 = carve((size_t)Nn * H2 * 4);  // h1 @ W2
    float* acc2   = carve((size_t)Nn * H2 * 4);  // layer-2 aggregate -> z
    (void)ws_size;

    const int TB = 256;
    auto nblk = [](long n, int tb) { return (int)((n + tb - 1) / tb); };

    // Init: self-loop gives every node degree 1; accumulators start at zero.
    fill_f32_kernel<<<nblk(Nn, TB), TB, 0, stream>>>(dinv, 1.0f, Nn);
    fill_f32_kernel<<<nblk((long)Nn * H1, TB), TB, 0, stream>>>(acc1, 0.0f, Nn * H1);
    fill_f32_kernel<<<nblk((long)Nn * H2, TB), TB, 0, stream>>>(acc2, 0.0f, Nn * H2);

    // Degree + normalization (shared by both layers).
    deg_kernel<<<nblk(E, TB), TB, 0, stream>>>(dst, E, dinv);
    rsqrt_kernel<<<nblk(Nn, TB), TB, 0, stream>>>(dinv, Nn);

    // Each block covers 4 wave-owned 16-row tiles -> 64 rows.
    const int gemm_blocks = (Nn + 63) / 64;

    // Layer 1: WMMA transform (N=50 -> NT=4) -> scatter -> self+bias+ReLU.
    wmma_gemm_kernel<4><<<gemm_blocks, 128, 0, stream>>>(x, W1, h_lin1, Nn, F_IN, H1);
    scatter_kernel<<<nblk(E, TB), TB, 0, stream>>>(h_lin1, src, dst, dinv, acc1, E, H1);
    finish_kernel<<<nblk((long)Nn * H1, TB), TB, 0, stream>>>(h_lin1, dinv, b1, acc1,
                                                              Nn, H1, /*relu=*/1);

    // Layer 2: WMMA transform (N=10 -> NT=1) -> scatter -> self+bias.
    wmma_gemm_kernel<1><<<gemm_blocks, 128, 0, stream>>>(acc1, W2, h_lin2, Nn, H1, H2);
    scatter_kernel<<<nblk(E, TB), TB, 0, stream>>>(h_lin2, src, dst, dinv, acc2, E, H2);
    finish_kernel<<<nblk((long)Nn * H2, TB), TB, 0, stream>>>(h_lin2, dinv, b2, acc2,
                                                              Nn, H2, /*relu=*/0);

    // Decode: edge-wise dot products over concat(pos, neg).
    decode_kernel<<<nblk((long)2 * EP, TB), TB, 0, stream>>>(
        acc2, pei, pei + EP, nei, nei + EP, (float*)d_out, EP, H2);
}